// GTREncoder_6433861009964
// MI455X (gfx1250) — compile-verified
//
#include <hip/hip_runtime.h>
#include <hip/hip_bf16.h>

typedef __attribute__((ext_vector_type(2))) float v2f;
typedef __attribute__((ext_vector_type(8))) float v8f;
typedef __attribute__((ext_vector_type(4))) unsigned int v4u;
typedef __attribute__((ext_vector_type(8))) int v8i;
typedef __attribute__((ext_vector_type(4))) int v4i;

#define EMB   128
#define HID   256
#define NB    64
#define NT    512
#define NG    1024          // 4*HID
#define BSTRIDE 260         // padded LDS row stride (floats) -> conflict-free

#if __has_builtin(__builtin_amdgcn_tensor_load_to_lds) && \
    __has_builtin(__builtin_amdgcn_s_wait_tensorcnt)
#define HAVE_TDM 1
#else
#define HAVE_TDM 0
#endif

__device__ __forceinline__ float sigf(float x) { return 1.0f / (1.0f + __expf(-x)); }

// ---------------------------------------------------------------------------
// K0: x[row] = [mean(emb[node[row,0..1]]), mean(emb[rel[row,0..2]])], row = b*T+t
//     also fclamp[row] = min(father[row], t-1)
// ---------------------------------------------------------------------------
__global__ __launch_bounds__(128) void embed_kernel(
    const int* __restrict__ node, const int* __restrict__ rel,
    const int* __restrict__ father, const float* __restrict__ emb,
    float* __restrict__ x, int* __restrict__ fclamp) {
  const int row = blockIdx.x;            // b*NT + t
  const int e   = threadIdx.x;           // 0..127
  const int t   = row & (NT - 1);
  const int n0 = node[row * 2 + 0], n1 = node[row * 2 + 1];
  const int r0 = rel[row * 3 + 0], r1 = rel[row * 3 + 1], r2 = rel[row * 3 + 2];
  float ne = (emb[(size_t)n0 * EMB + e] + emb[(size_t)n1 * EMB + e]) * 0.5f;
  float re = (emb[(size_t)r0 * EMB + e] + emb[(size_t)r1 * EMB + e] +
              emb[(size_t)r2 * EMB + e]) * (1.0f / 3.0f);
  x[(size_t)row * (2 * EMB) + e]        = ne;
  x[(size_t)row * (2 * EMB) + EMB + e]  = re;
  if (e == 0) fclamp[row] = min(father[row], t - 1);
}

// ---------------------------------------------------------------------------
// Generic fp32 WMMA GEMM: C[M,N] = A[M,K] @ W[N,K]^T (+ bias[n]).
// One wave per 16x16 C tile, V_WMMA_F32_16X16X4_F32 over K.
// ---------------------------------------------------------------------------
__global__ __launch_bounds__(256) void wmma_gemm_xwt(
    const float* __restrict__ A, const float* __restrict__ W,
    const float* __restrict__ bias, float* __restrict__ C,
    int M, int N, int K) {
  const int wid  = blockIdx.x * 8 + (threadIdx.x >> 5);
  const int lane = threadIdx.x & 31;
  const int half = lane >> 4, ln = lane & 15;
  const int ntiles = N >> 4;
  const int mt = wid / ntiles, nt = wid - mt * ntiles;
  const int m0 = mt << 4, n0 = nt << 4;

  const float* Arow = A + (size_t)(m0 + ln) * K + half * 2;
  const float* Wrow = W + (size_t)(n0 + ln) * K + half * 2;

  v8f acc = {};
  for (int k0 = 0; k0 < K; k0 += 4) {
    v2f a = *(const v2f*)(Arow + k0);
    v2f b = *(const v2f*)(Wrow + k0);
    acc = __builtin_amdgcn_wmma_f32_16x16x4_f32(false, a, false, b,
                                                (short)0, acc, false, false);
  }
  const float bv = bias ? bias[n0 + ln] : 0.0f;
#pragma unroll
  for (int v = 0; v < 8; ++v) {
    const int m = m0 + v + half * 8;
    C[(size_t)m * N + n0 + ln] = acc[v] + bv;
  }
}

// ---------------------------------------------------------------------------
// K2: persistent tree-LSTM scan. 16 blocks (one 16-wide j slice each),
// 4 waves/block (one 16-batch tile each). W_hh slice staged in LDS via the
// Tensor Data Mover (3D tile, LDS row-padding -> conflict-free stride 260).
// Per step: gx prefetch, 4 gate tiles x 64 v_wmma_f32_16x16x4_f32 (K=256),
// LSTM activations, then device-scope atomic barrier.
// ---------------------------------------------------------------------------
__global__ __launch_bounds__(128) void scan_kernel(
    const float* __restrict__ gx, const float* __restrict__ W_hh,
    const float* __restrict__ b_hh, const int* __restrict__ fclamp,
    float* __restrict__ h_mem, float* __restrict__ c_mem,
    unsigned* __restrict__ ctr) {
  extern __shared__ float bsm[];                 // 64 rows * BSTRIDE floats
  const int jt   = blockIdx.x;                   // 0..15 : hidden-j tile
  const int lane = threadIdx.x & 31;
  const int half = lane >> 4, ln = lane & 15;
  const int b0   = (threadIdx.x >> 5) << 4;      // batch-tile origin (0,16,32,48)

#if HAVE_TDM
  // --- TDM: one tensor_load_to_lds moves the whole 64x256 weight slice.
  // Tile: x=256 floats (contiguous), y=16 rows (stride 256), z=4 gates
  // (stride HID*HID). pad_interval=7 (256 DW), pad_amount=3 (4 DW) => the
  // LDS rows land at stride 260 floats, matching the ds_load_b64 swizzle.
  if (threadIdx.x < 32) {                        // wave 0 issues the DMA
    const unsigned long long ga =
        (unsigned long long)(uintptr_t)(W_hh + (size_t)jt * 16 * HID);
    const unsigned ldsoff = (unsigned)(uintptr_t)(&bsm[0]);
    v4u g0;
    g0.x = 1u;                                           // count=1 (valid)
    g0.y = ldsoff;                                       // lds_addr
    g0.z = (unsigned)(ga & 0xffffffffu);                 // global_addr[31:0]
    g0.w = (unsigned)((ga >> 32) & 0x01ffffffu) | 0x80000000u; // addr[56:32] | type=2
    v8i g1;
    g1[0] = (2 << 16) | (1 << 20) | (7 << 22) | (3 << 25); // 4B, pad on, 256DW/4DW
    g1[1] = (int)((unsigned)(HID & 0xffff) << 16);         // tensor_dim0=256 (lo16)
    g1[2] = (int)(16u << 16);                              // tensor_dim1=16
    g1[3] = (int)((unsigned)(HID & 0xffff) << 16);         // tile_dim0=256
    g1[4] = 16 | (4 << 16);                                // tile_dim1=16, tile_dim2=4
    g1[5] = HID;                                           // tensor_dim0_stride=256
    g1[6] = 0;                                             // strides hi / dim1_stride lo
    g1[7] = 1;                                             // tensor_dim1_stride=65536 (hi32)
    v4i g2 = {};
    g2.x = 4;                                              // tensor_dim2=4
    v4i g3 = {};
#if defined(__clang_major__) && (__clang_major__ >= 23)
    v8i g4 = {};
    __builtin_amdgcn_tensor_load_to_lds(g0, g1, g2, g3, g4, 0);
#else
    __builtin_amdgcn_tensor_load_to_lds(g0, g1, g2, g3, 0);
#endif
    __builtin_amdgcn_s_wait_tensorcnt(0);
  }
#else
  for (int idx = threadIdx.x; idx < 64 * HID; idx += 128) {
    const int lr = idx >> 8;
    const int k  = idx & (HID - 1);
    const int g  = lr >> 4, r = lr & 15;
    bsm[lr * BSTRIDE + k] = W_hh[(size_t)(g * HID + jt * 16 + r) * HID + k];
  }
#endif
  __syncthreads();

  const int j = jt * 16 + ln;                    // this lane's column j
  const float bh0 = b_hh[0 * HID + j];
  const float bh1 = b_hh[1 * HID + j];
  const float bh2 = b_hh[2 * HID + j];
  const float bh3 = b_hh[3 * HID + j];

  for (int t = 0; t < NT; ++t) {
    // ---- prefetch this step's gx segments (hide HBM latency behind WMMAs)
    {
      const size_t gxo = ((size_t)(b0 + ln) * NT + t) * NG + jt * 16;
      __builtin_prefetch(gx + gxo + 0 * HID, 0, 0);
      __builtin_prefetch(gx + gxo + 1 * HID, 0, 0);
      __builtin_prefetch(gx + gxo + 2 * HID, 0, 0);
      __builtin_prefetch(gx + gxo + 3 * HID, 0, 0);
    }

    // ---- A operand: gathered h_prev row for batch bA = b0 + ln
    const int bA = b0 + ln;
    const int f  = fclamp[bA * NT + t];
    const float* hrow = h_mem + ((size_t)(f + 1) * NB + bA) * HID + half * 2;

    v8f acc[4] = {};
    for (int k0 = 0; k0 < HID; k0 += 4) {
      v2f a = *(const v2f*)(hrow + k0);
#pragma unroll
      for (int g = 0; g < 4; ++g) {
        v2f b = *(const v2f*)&bsm[(g * 16 + ln) * BSTRIDE + k0 + half * 2];
        acc[g] = __builtin_amdgcn_wmma_f32_16x16x4_f32(false, a, false, b,
                                                       (short)0, acc[g],
                                                       false, false);
      }
    }

    // ---- activations: element (v) of C maps to batch b = b0 + v + 8*half
#pragma unroll
    for (int v = 0; v < 8; ++v) {
      const int b = b0 + v + half * 8;
      const size_t gxo = ((size_t)b * NT + t) * NG;
      const float gi = acc[0][v] + gx[gxo + 0 * HID + j] + bh0;
      const float gf = acc[1][v] + gx[gxo + 1 * HID + j] + bh1;
      const float gg = acc[2][v] + gx[gxo + 2 * HID + j] + bh2;
      const float go = acc[3][v] + gx[gxo + 3 * HID + j] + bh3;
      const int  fb = fclamp[b * NT + t];
      const float cprev = c_mem[((size_t)(fb + 1) * NB + b) * HID + j];
      const float cn = sigf(gf) * cprev + sigf(gi) * tanhf(gg);
      const float hn = sigf(go) * tanhf(cn);
      h_mem[((size_t)(t + 1) * NB + b) * HID + j] = hn;
      c_mem[((size_t)(t + 1) * NB + b) * HID + j] = cn;
    }

    // ---- device-scope barrier (monotone target; ctr zeroed per launch)
    __syncthreads();
    if (threadIdx.x == 0) {
      __hip_atomic_fetch_add(ctr, 1u, __ATOMIC_RELEASE, __HIP_MEMORY_SCOPE_AGENT);
      const unsigned target = (unsigned)(t + 1) * gridDim.x;
      while (__hip_atomic_load(ctr, __ATOMIC_ACQUIRE, __HIP_MEMORY_SCOPE_AGENT)
             < target) {
        __builtin_amdgcn_s_sleep(1);
      }
    }
    __syncthreads();
  }
}

// ---------------------------------------------------------------------------
// K3a: encoder_outputs[b,t,j] = h_mem[t+1,b,j] * mask[b,t]
// ---------------------------------------------------------------------------
__global__ __launch_bounds__(256) void mask_out_kernel(
    const float* __restrict__ h_mem, const float* __restrict__ mask,
    float* __restrict__ out0) {
  const size_t idx = (size_t)blockIdx.x * 256 + threadIdx.x;  // (b*NT+t)*HID + j
  const int j = (int)(idx & (HID - 1));
  const int bt = (int)(idx >> 8);                              // b*NT + t
  const int t = bt & (NT - 1);
  const int b = bt >> 9;
  out0[idx] = h_mem[((size_t)(t + 1) * NB + b) * HID + j] * mask[bt];
}

// ---------------------------------------------------------------------------
// K3c: masked max over T of h_mem / c_mem -> h_max, c_max (B*HID each)
// ---------------------------------------------------------------------------
__global__ __launch_bounds__(256) void max_kernel(
    const float* __restrict__ h_mem, const float* __restrict__ c_mem,
    const float* __restrict__ mask, float* __restrict__ hmax_out,
    float* __restrict__ cmax_out) {
  const int id = blockIdx.x * 256 + threadIdx.x;   // 0..NB*HID-1
  const int b = id >> 8, j = id & (HID - 1);
  float hm = -INFINITY, cm = -INFINITY;
  for (int t = 0; t < NT; ++t) {
    const float m = mask[b * NT + t];
    const size_t o = ((size_t)(t + 1) * NB + b) * HID + j;
    hm = fmaxf(hm, h_mem[o] * m);
    cm = fmaxf(cm, c_mem[o] * m);
  }
  hmax_out[id] = hm;
  cmax_out[id] = cm;
}

// ---------------------------------------------------------------------------
extern "C" void kernel_launch(void* const* d_in, const int* in_sizes, int n_in,
                              void* d_out, int out_size, void* d_ws, size_t ws_size,
                              hipStream_t stream) {
  const int*   node   = (const int*)d_in[0];
  const int*   rel    = (const int*)d_in[1];
  const int*   father = (const int*)d_in[2];
  const float* mask   = (const float*)d_in[3];
  const float* emb    = (const float*)d_in[4];
  const float* W_ih   = (const float*)d_in[5];
  const float* W_hh   = (const float*)d_in[6];
  const float* b_ih   = (const float*)d_in[7];
  const float* b_hh   = (const float*)d_in[8];
  const float* W_h    = (const float*)d_in[9];

  char* ws = (char*)d_ws;
  size_t off = 0;
  float* x     = (float*)(ws + off); off += (size_t)NB * NT * (2 * EMB) * 4;   // 33.5 MB
  float* gx    = (float*)(ws + off); off += (size_t)NB * NT * NG * 4;          // 134 MB
  float* h_mem = (float*)(ws + off); off += (size_t)(NT + 1) * NB * HID * 4;   // 33.6 MB
  float* c_mem = (float*)(ws + off); off += (size_t)(NT + 1) * NB * HID * 4;   // 33.6 MB
  int*   fcl   = (int*)(ws + off);   off += (size_t)NB * NT * 4;
  unsigned* ctr = (unsigned*)(ws + off); off += 64;

  float* out0 = (float*)d_out;                        // encoder_outputs (B,T,H)
  float* out1 = out0 + (size_t)NB * NT * HID;         // encoder_feature (B*T,H)
  float* out2 = out1 + (size_t)NB * NT * HID;         // h_max (1,B,H)
  float* out3 = out2 + (size_t)NB * HID;              // c_max (1,B,H)

  // Per-launch state init (graph-capture safe, replay-deterministic).
  hipMemsetAsync(ctr, 0, sizeof(unsigned), stream);
  hipMemsetAsync(h_mem, 0, (size_t)NB * HID * 4, stream);   // h_mem[0] = 0
  hipMemsetAsync(c_mem, 0, (size_t)NB * HID * 4, stream);   // c_mem[0] = 0

  // Phase 1: embeddings + gx = x @ W_ih^T + b_ih
  embed_kernel<<<NB * NT, 128, 0, stream>>>(node, rel, father, emb, x, fcl);
  {
    const int tiles = (NB * NT / 16) * (NG / 16);   // 2048 * 64
    wmma_gemm_xwt<<<tiles / 8, 256, 0, stream>>>(x, W_ih, b_ih, gx,
                                                 NB * NT, NG, 2 * EMB);
  }

  // Phase 2: serial tree-LSTM scan (persistent, 16 blocks, atomic barrier)
  scan_kernel<<<16, 128, 64 * BSTRIDE * 4, stream>>>(gx, W_hh, b_hh, fcl,
                                                     h_mem, c_mem, ctr);

  // Phase 3: masked outputs, feature GEMM, masked max reductions
  mask_out_kernel<<<(NB * NT * HID) / 256, 256, 0, stream>>>(h_mem, mask, out0);
  {
    const int tiles = (NB * NT / 16) * (HID / 16);  // 2048 * 16
    wmma_gemm_xwt<<<tiles / 8, 256, 0, stream>>>(out0, W_h, nullptr, out1,
                                                 NB * NT, HID, HID);
  }
  max_kernel<<<(NB * HID) / 256, 256, 0, stream>>>(h_mem, c_mem, mask, out2, out3);
}